// RNN_27685359190558
// MI455X (gfx1250) — compile-verified
//
#include <hip/hip_runtime.h>
#include <math.h>

typedef __attribute__((ext_vector_type(2))) float v2f;
typedef __attribute__((ext_vector_type(8))) float v8f;

#define T_STEPS 28
#define F_IN    28
#define H_UNITS 128
#define G4      512   // 4*H
#define TILE_M  32    // batch rows per workgroup (2 WMMA M-tiles)
#define NKF_U   32    // K fragments for h@U (128/4)
#define NKF_W   7     // K fragments for x@W (28/4)

__device__ __forceinline__ float fast_sigmoid(float x) {
    return 1.0f / (1.0f + __expf(-x));
}

// One workgroup = 256 threads (8 wave32) owns a 32-row batch tile for all T steps.
// Wave w owns hidden units [16w, 16w+16): it computes the four gate column tiles
// {n, n+128, n+256, n+384} so the LSTM cell update is register-resident (no z spill,
// c-state lives in VGPRs). Only h round-trips LDS (needed as next-step A by all waves).
//
// LDS: uSwz 128*512 floats (256KB) + hSwz 32*128 (16KB) = 272KB of the 320KB WGP LDS.
//
// uSwz is per-wave contiguous so all B-fragment ds_load_b64 use immediate offsets:
//   uSwz[w*8192 + kf*256 + g*64 + laneN*4 + half*2 + j] = U[(4kf+2half+j)*512 + g*128 + w*16 + laneN]
//   -> per (kf,g) load: byte offset kf*1024 + g*256 (<= 32512, fits DS imm16),
//      banks (laneN*4 + half*2) % 64 cover all 64 -> conflict-free b64.
// hSwz (A fragments), k = 4*kf + 2*half + j:
//   hSwz[kf*128 + m*4 + half*2 + j] = h[m][k]   (16KB window, imm offsets too)
__global__ __launch_bounds__(256, 1)
void lstm_fused_wmma3(const float* __restrict__ x, const float* __restrict__ W,
                      const float* __restrict__ U, const float* __restrict__ b,
                      const float* __restrict__ Wd, const float* __restrict__ bd,
                      float* __restrict__ out)
{
    extern __shared__ float smem[];
    float* uSwz = smem;                    // 65536 floats
    float* hSwz = smem + 65536;            // 4096 floats

    const int tid   = threadIdx.x;
    const int lane  = tid & 31;
    const int wave  = tid >> 5;
    const int laneN = lane & 15;           // N col (B/C/D) or M row (A) within a 16-tile
    const int half  = lane >> 4;           // selects K pair within a K=4 fragment
    const int rowBase = blockIdx.x * TILE_M;

    // ---- Stage U into LDS, per-wave-contiguous B-fragment layout ----
    for (int i = tid; i < H_UNITS * G4; i += 256) {
        const int j  = i & 1;
        const int hf = (i >> 1) & 1;
        const int ln = (i >> 2) & 15;
        const int g  = (i >> 6) & 3;
        const int kf = (i >> 8) & 31;
        const int w  = i >> 13;
        uSwz[i] = U[(kf * 4 + hf * 2 + j) * G4 + (g * H_UNITS + w * 16 + ln)];
    }
    // h0 = 0
    for (int i = tid; i < TILE_M * H_UNITS; i += 256) hSwz[i] = 0.0f;
    __syncthreads();

    // This wave's hidden unit for the gate math: n = 16*wave + laneN
    const int nUnit = wave * 16 + laneN;
    const int nOff  = (nUnit >> 2) * 128 + ((nUnit >> 1) & 1) * 2 + (nUnit & 1);

    // Per-lane base pointers (all hot-loop DS offsets become immediates)
    const float* uW  = uSwz + wave * 8192 + laneN * 4 + half * 2;
    const float* hA0 = hSwz + laneN * 4 + half * 2;
    const float* hA1 = hSwz + (16 + laneN) * 4 + half * 2;

    const float* xRow0 = x + (size_t)(rowBase + laneN) * (T_STEPS * F_IN);
    const float* xRow1 = x + (size_t)(rowBase + 16 + laneN) * (T_STEPS * F_IN);

    // Persistent bias broadcast vectors: C operand of the first WMMA in each chain
    v8f biasV[4];
    #pragma unroll
    for (int g = 0; g < 4; ++g) {
        const float bias = b[g * H_UNITS + wave * 16 + laneN];
        biasV[g] = (v8f){bias,bias,bias,bias,bias,bias,bias,bias};
    }

    // Persistent cell state in registers
    v8f cSt[2];
    cSt[0] = (v8f){0.f,0.f,0.f,0.f,0.f,0.f,0.f,0.f};
    cSt[1] = cSt[0];

    for (int t = 0; t < T_STEPS; ++t) {
        v8f acc[4][2];

        // ---- z = bias + x_t @ W  (K = 28 -> 7 fragments) ; W from global (L2) ----
        // kf = 0 seeds the accumulators with biasV as the C operand.
        {
            const int k = half * 2;
            v2f a0, a1;
            a0.x = xRow0[t * F_IN + k];  a0.y = xRow0[t * F_IN + k + 1];
            a1.x = xRow1[t * F_IN + k];  a1.y = xRow1[t * F_IN + k + 1];
            #pragma unroll
            for (int g = 0; g < 4; ++g) {
                const int col = g * H_UNITS + wave * 16 + laneN;
                v2f bb;
                bb.x = W[k * G4 + col];
                bb.y = W[(k + 1) * G4 + col];
                acc[g][0] = __builtin_amdgcn_wmma_f32_16x16x4_f32(
                    false, a0, false, bb, (short)0, biasV[g], false, false);
                acc[g][1] = __builtin_amdgcn_wmma_f32_16x16x4_f32(
                    false, a1, false, bb, (short)0, biasV[g], false, false);
            }
        }
        #pragma unroll
        for (int kf = 1; kf < NKF_W; ++kf) {
            const int k = kf * 4 + half * 2;
            v2f a0, a1;
            a0.x = xRow0[t * F_IN + k];  a0.y = xRow0[t * F_IN + k + 1];
            a1.x = xRow1[t * F_IN + k];  a1.y = xRow1[t * F_IN + k + 1];
            #pragma unroll
            for (int g = 0; g < 4; ++g) {
                const int col = g * H_UNITS + wave * 16 + laneN;
                v2f bb;
                bb.x = W[k * G4 + col];
                bb.y = W[(k + 1) * G4 + col];
                acc[g][0] = __builtin_amdgcn_wmma_f32_16x16x4_f32(
                    false, a0, false, bb, (short)0, acc[g][0], false, false);
                acc[g][1] = __builtin_amdgcn_wmma_f32_16x16x4_f32(
                    false, a1, false, bb, (short)0, acc[g][1], false, false);
            }
        }

        // ---- z += h @ U  (K = 128 -> 32 fragments) ; imm-offset b64 LDS loads ----
        for (int kf = 0; kf < NKF_U; ++kf) {
            const v2f a0 = *(const v2f*)&hA0[kf * 128];
            const v2f a1 = *(const v2f*)&hA1[kf * 128];
            #pragma unroll
            for (int g = 0; g < 4; ++g) {
                const v2f bb = *(const v2f*)&uW[kf * 256 + g * 64];
                acc[g][0] = __builtin_amdgcn_wmma_f32_16x16x4_f32(
                    false, a0, false, bb, (short)0, acc[g][0], false, false);
                acc[g][1] = __builtin_amdgcn_wmma_f32_16x16x4_f32(
                    false, a1, false, bb, (short)0, acc[g][1], false, false);
            }
        }

        __syncthreads();   // all waves finished reading hSwz for this step

        // ---- LSTM cell update, fully register-resident (Keras order i,f,g,o) ----
        #pragma unroll
        for (int mt = 0; mt < 2; ++mt) {
            #pragma unroll
            for (int r = 0; r < 8; ++r) {
                const float ig = fast_sigmoid(acc[0][mt][r]);
                const float fg = fast_sigmoid(acc[1][mt][r]);
                const float gg = fmaxf(acc[2][mt][r], 0.0f);   // relu candidate
                const float og = fast_sigmoid(acc[3][mt][r]);
                const float cn = fg * cSt[mt][r] + ig * gg;
                cSt[mt][r] = cn;
                const float hv = og * fmaxf(cn, 0.0f);         // relu cell state
                const int m = mt * 16 + r + 8 * half;          // C-layout row
                hSwz[nOff + m * 4] = hv;
            }
        }

        __syncthreads();   // h stores visible before next step's A reads
    }

    // ---- Dense (128 -> 10) + softmax; threads 0..31, one batch row each ----
    if (tid < TILE_M) {
        const int m = tid;
        float logits[10];
        #pragma unroll
        for (int c = 0; c < 10; ++c) logits[c] = bd[c];
        for (int k = 0; k < H_UNITS; ++k) {
            const float hv = hSwz[(k >> 2) * 128 + m * 4 + ((k >> 1) & 1) * 2 + (k & 1)];
            #pragma unroll
            for (int c = 0; c < 10; ++c) logits[c] += hv * Wd[k * 10 + c];
        }
        float mx = logits[0];
        #pragma unroll
        for (int c = 1; c < 10; ++c) mx = fmaxf(mx, logits[c]);
        float e[10], sum = 0.0f;
        #pragma unroll
        for (int c = 0; c < 10; ++c) { e[c] = __expf(logits[c] - mx); sum += e[c]; }
        const float inv = 1.0f / sum;
        #pragma unroll
        for (int c = 0; c < 10; ++c)
            out[(size_t)(rowBase + m) * 10 + c] = e[c] * inv;
    }
}

extern "C" void kernel_launch(void* const* d_in, const int* in_sizes, int n_in,
                              void* d_out, int out_size, void* d_ws, size_t ws_size,
                              hipStream_t stream) {
    const float* x  = (const float*)d_in[0];
    const float* W  = (const float*)d_in[1];
    const float* U  = (const float*)d_in[2];
    const float* b  = (const float*)d_in[3];
    const float* Wd = (const float*)d_in[4];
    const float* bd = (const float*)d_in[5];
    float* out = (float*)d_out;

    const size_t ldsBytes = (size_t)(65536 + 4096) * sizeof(float); // 272 KB
    const int numBlocks = 16384 / TILE_M; // 512

    hipLaunchKernelGGL(lstm_fused_wmma3, dim3(numBlocks), dim3(256), ldsBytes, stream,
                       x, W, U, b, Wd, bd, out);

    (void)in_sizes; (void)n_in; (void)out_size; (void)d_ws; (void)ws_size;
}